// FeaturePropagation_14061722927194
// MI455X (gfx1250) — compile-verified
//
#include <hip/hip_runtime.h>
#include <stdint.h>

// ---------------------------------------------------------------------------
// FeaturePropagation on MI455X (gfx1250, wave32)
//
// Roofline: 4.1 GFLOP total vs ~9.4 GB of (L2-resident) traffic -> purely
// memory bound; WMMA cannot help (no dense tiles, AI ~0.44 flop/B). Strategy:
//  * one-time CSR build -> zero atomics in the 20-iteration hot loop
//  * wave-per-node gather/accumulate, float4 per lane (512 B coalesced rows)
//  * gfx1250 async global->LDS gather (ASYNCcnt-tracked) for the dominant
//    410 MB/iter random row gather; split-counter waits for sync
// ---------------------------------------------------------------------------

#define F 128                 // features per node (reference: D_FEAT)
#define NUM_ITERS 20          // reference: NUM_ITERATIONS
#define CHUNK 16              // edges staged per wave per step
#define WAVES_PER_BLOCK 8
#define TPB (WAVES_PER_BLOCK * 32)

#if __has_builtin(__builtin_amdgcn_global_load_async_to_lds_b128)
#define USE_ASYNC_GATHER 1
#else
#define USE_ASYNC_GATHER 0
#endif

// Exact pointer types the async builtin expects (from clang's diagnostic):
// param0: v4i in addrspace(1) (printed as "__device__"), param1: v4i in LDS.
typedef int v4i __attribute__((ext_vector_type(4)));
typedef __attribute__((address_space(1))) v4i* as1_v4i_p;
typedef __attribute__((address_space(3))) v4i* as3_v4i_p;

__device__ __forceinline__ void wait_asynccnt0() {
#if __has_builtin(__builtin_amdgcn_s_wait_asynccnt)
  __builtin_amdgcn_s_wait_asynccnt(0);
#elif defined(__AMDGCN__)
  asm volatile("s_wait_asynccnt 0" ::: "memory");
#endif
}

__device__ __forceinline__ void wait_dscnt0() {
#if __has_builtin(__builtin_amdgcn_s_wait_dscnt)
  __builtin_amdgcn_s_wait_dscnt(0);
#elif defined(__AMDGCN__)
  asm volatile("s_wait_dscnt 0" ::: "memory");
#endif
}

// ---------------------------------------------------------------------------
// CSR construction (once per launch; amortized over 20 SpMM iterations)
// ---------------------------------------------------------------------------

__global__ void zero_i32_kernel(int* __restrict__ p, int n) {
  int i = blockIdx.x * blockDim.x + threadIdx.x;
  if (i < n) p[i] = 0;
}

__global__ void count_deg_kernel(const long long* __restrict__ row,
                                 int* __restrict__ deg, int e_count) {
  int e = blockIdx.x * blockDim.x + threadIdx.x;
  if (e < e_count) atomicAdd(&deg[(int)row[e]], 1);
}

// Single-block exclusive scan over N degrees -> rowptr[N+1]; also seeds cursor.
__global__ void scan_deg_kernel(const int* __restrict__ deg,
                                int* __restrict__ rowptr,
                                int* __restrict__ cursor, int n) {
  __shared__ int part[257];
  const int t = threadIdx.x;                   // blockDim.x == 256, grid == 1
  const int chunk = (n + 255) / 256;
  const int s0 = t * chunk;
  const int s1 = min(s0 + chunk, n);
  int s = 0;
  for (int i = s0; i < s1; ++i) s += deg[i];
  part[t] = s;
  __syncthreads();
  if (t == 0) {
    int run = 0;
    for (int i = 0; i < 256; ++i) { int v = part[i]; part[i] = run; run += v; }
    part[256] = run;
  }
  __syncthreads();
  int run = part[t];
  for (int i = s0; i < s1; ++i) {
    rowptr[i] = run;
    cursor[i] = run;
    run += deg[i];
  }
  if (t == 0) rowptr[n] = part[256];
}

__global__ void fill_csr_kernel(const long long* __restrict__ row,
                                const long long* __restrict__ col,
                                const float* __restrict__ w,
                                int* __restrict__ cursor,
                                int* __restrict__ col_s,
                                float* __restrict__ w_s, int e_count) {
  int e = blockIdx.x * blockDim.x + threadIdx.x;
  if (e < e_count) {
    int r = (int)row[e];
    int p = atomicAdd(&cursor[r], 1);
    col_s[p] = (int)col[e];
    w_s[p] = w[e];
  }
}

// ---------------------------------------------------------------------------
// Fused propagate iteration: nxt[i] = mask(x[i]) ? x[i] : sum_e w_e*cur[col_e]
// One wave per node; lane L owns features [4L, 4L+4) as a float4.
// ---------------------------------------------------------------------------

__global__ __launch_bounds__(TPB) void propagate_kernel(
    const float* __restrict__ cur, float* __restrict__ nxt,
    const float* __restrict__ x, const int* __restrict__ rowptr,
    const int* __restrict__ cols, const float* __restrict__ wts, int n) {
#if USE_ASYNC_GATHER
  __shared__ float stage[WAVES_PER_BLOCK][CHUNK][F];   // 64 KB / workgroup
  const int wv = threadIdx.x >> 5;
#endif
  const int wave = (int)((blockIdx.x * (unsigned)TPB + threadIdx.x) >> 5);
  const int lane = threadIdx.x & 31;
  if (wave >= n) return;

  const int begin = rowptr[wave];
  const int end = rowptr[wave + 1];

  float4 acc; acc.x = 0.f; acc.y = 0.f; acc.z = 0.f; acc.w = 0.f;

  for (int j = begin; j < end; j += CHUNK) {
    const int m = min(CHUNK, end - j);
    int c = 0; float w = 0.f;
    if (lane < m) {                    // lanes 0..m-1 carry this chunk's edges
      c = cols[j + lane];
      w = wts[j + lane];
    }
#if USE_ASYNC_GATHER
    // Previous chunk's LDS reads must drain before the async engine rewrites
    // the stage buffer (DS pipe and ASYNC pipe are mutually unordered).
    wait_dscnt0();
    for (int k = 0; k < m; ++k) {
      const int ck = __shfl(c, k, 32);
      const float* gsrc = cur + (size_t)ck * F + lane * 4;   // 16 B per lane
      float* ldst = &stage[wv][k][lane * 4];
      __builtin_amdgcn_global_load_async_to_lds_b128(
          (as1_v4i_p)(uintptr_t)gsrc,
          (as3_v4i_p)(uint32_t)(uintptr_t)ldst, 0, 0);
    }
    wait_asynccnt0();
    for (int k = 0; k < m; ++k) {
      const float wk = __shfl(w, k, 32);
      const float4 v = *(const float4*)&stage[wv][k][lane * 4];
      acc.x = fmaf(wk, v.x, acc.x);
      acc.y = fmaf(wk, v.y, acc.y);
      acc.z = fmaf(wk, v.z, acc.z);
      acc.w = fmaf(wk, v.w, acc.w);
    }
#else
    if (lane < m) __builtin_prefetch(cur + (size_t)c * F, 0, 0); // global_prefetch_b8
    for (int k = 0; k < m; ++k) {
      const int ck = __shfl(c, k, 32);
      const float wk = __shfl(w, k, 32);
      const float4 v = *(const float4*)(cur + (size_t)ck * F + lane * 4);
      acc.x = fmaf(wk, v.x, acc.x);
      acc.y = fmaf(wk, v.y, acc.y);
      acc.z = fmaf(wk, v.z, acc.z);
      acc.w = fmaf(wk, v.w, acc.w);
    }
#endif
  }

  const size_t base = (size_t)wave * F + lane * 4;
  const float4 xv = *(const float4*)(x + base);
  float4 o;
  o.x = (xv.x != 0.f) ? xv.x : acc.x;
  o.y = (xv.y != 0.f) ? xv.y : acc.y;
  o.z = (xv.z != 0.f) ? xv.z : acc.z;
  o.w = (xv.w != 0.f) ? xv.w : acc.w;
  *(float4*)(nxt + base) = o;
}

// ---------------------------------------------------------------------------
// Host-side orchestration (graph-capture safe: kernels only, all on `stream`)
// ---------------------------------------------------------------------------

extern "C" void kernel_launch(void* const* d_in, const int* in_sizes, int n_in,
                              void* d_out, int out_size, void* d_ws, size_t ws_size,
                              hipStream_t stream) {
  (void)n_in; (void)out_size; (void)ws_size;

  const float* x = (const float*)d_in[0];            // [N, 128] f32
  const long long* ei = (const long long*)d_in[1];   // [2, E] i64
  const float* ew = (const float*)d_in[2];           // [E] f32

  const int N = in_sizes[0] / F;
  const int E = in_sizes[2];
  const long long* row = ei;          // edge_index[0]
  const long long* col = ei + E;      // edge_index[1]

  // Workspace layout (~32.5 MB for N=50k, E=800k; fully rewritten each call)
  char* ws = (char*)d_ws;
  size_t off = 0;
  float* buf = (float*)(ws + off); off += (size_t)N * F * sizeof(float);
  off = (off + 255) & ~(size_t)255;
  int* deg = (int*)(ws + off);     off += (size_t)N * sizeof(int);
  off = (off + 255) & ~(size_t)255;
  int* rowptr = (int*)(ws + off);  off += (size_t)(N + 1) * sizeof(int);
  off = (off + 255) & ~(size_t)255;
  int* cursor = (int*)(ws + off);  off += (size_t)N * sizeof(int);
  off = (off + 255) & ~(size_t)255;
  int* col_s = (int*)(ws + off);   off += (size_t)E * sizeof(int);
  off = (off + 255) & ~(size_t)255;
  float* w_s = (float*)(ws + off); off += (size_t)E * sizeof(float);

  const int gN = (N + 255) / 256;
  const int gE = (E + 255) / 256;

  zero_i32_kernel<<<gN, 256, 0, stream>>>(deg, N);
  count_deg_kernel<<<gE, 256, 0, stream>>>(row, deg, E);
  scan_deg_kernel<<<1, 256, 0, stream>>>(deg, rowptr, cursor, N);
  fill_csr_kernel<<<gE, 256, 0, stream>>>(row, col, ew, cursor, col_s, w_s, E);

  const int gridP = (N + WAVES_PER_BLOCK - 1) / WAVES_PER_BLOCK;
  const float* curp = x;
  for (int it = 1; it <= NUM_ITERS; ++it) {
    float* nxtp = (it & 1) ? buf : (float*)d_out;   // iter 20 (even) -> d_out
    propagate_kernel<<<gridP, TPB, 0, stream>>>(curp, nxtp, x, rowptr,
                                                col_s, w_s, N);
    curp = nxtp;
  }
}